// SupervisedBCEWithGraphConsistency_83391085019848
// MI455X (gfx1250) — compile-verified
//
#include <hip/hip_runtime.h>
#include <hip/hip_bf16.h>
#include <stdint.h>

// Problem constants (match reference: B=16, NB=512, BS=128, K=32)
#define B_  16
#define NB_ 512
#define BS_ 128
#define K_  32
#define N_  (NB_ * BS_)
#define GRAPH_WEIGHT_ 0.3f

// ---------------------------------------------------------------------------
// CDNA5 feature probes (compile-time)
// ---------------------------------------------------------------------------
#if defined(__gfx1250__) && __has_builtin(__builtin_amdgcn_global_load_async_to_lds_b32)
#define ASYNC_LDS 1
#else
#define ASYNC_LDS 0
#endif

#if defined(__gfx1250__) && __has_builtin(__builtin_amdgcn_wmma_f32_16x16x4_f32)
#define WMMA_RED 1
typedef __attribute__((ext_vector_type(2))) float v2f;
typedef __attribute__((ext_vector_type(8))) float v8f;
#else
#define WMMA_RED 0
#endif

// wave32 shuffle reduction (MI455X is wave32-only; do NOT assume 64 lanes)
__device__ __forceinline__ float wave_red_shfl(float v) {
    v += __shfl_xor(v, 16, 32);
    v += __shfl_xor(v, 8, 32);
    v += __shfl_xor(v, 4, 32);
    v += __shfl_xor(v, 2, 32);
    v += __shfl_xor(v, 1, 32);
    return v;
}

// Wave32 sum via the matrix pipe: D = A(16x4) * B(4x16) + 0 with
// A = {v twice per lane} (so A's 64 entries are exactly two copies of each
// lane's value, independent of the internal striping), B = all ones.
// Then D[m][n] = rowsum_A[m] for every column n. A lane's 8 D registers hold
// rows 0..7 (lanes 0-15) or rows 8..15 (lanes 16-31) of one column, so
// t = sum(d[0..7]) is the half-sum; t + shfl_xor(t,16) = 2*sum(v).
// EXEC must be all ones at call sites (it is: full blocks, no divergence).
__device__ __forceinline__ float wave_red_add(float v) {
#if WMMA_RED
    v2f a; a[0] = v;    a[1] = v;
    v2f b; b[0] = 1.0f; b[1] = 1.0f;
    v8f c = {};
    v8f d = __builtin_amdgcn_wmma_f32_16x16x4_f32(
        /*neg_a=*/false, a, /*neg_b=*/false, b,
        /*c_mod=*/(short)0, c, /*reuse_a=*/false, /*reuse_b=*/false);
    float t = ((d[0] + d[1]) + (d[2] + d[3])) + ((d[4] + d[5]) + (d[6] + d[7]));
    t += __shfl_xor(t, 16, 32);
    return 0.5f * t;
#else
    return wave_red_shfl(v);
#endif
}

__device__ __forceinline__ float sigmoidf_(float x) {
    return 1.0f / (1.0f + expf(-x));
}

// ---------------------------------------------------------------------------
// Kernel 0: zero the 34 accumulator floats (bce_sum, bce_cnt, loss_b[16], nunc_b[16])
// ---------------------------------------------------------------------------
__global__ void zero_acc_kernel(float* __restrict__ acc) {
    int t = threadIdx.x;
    if (t < 34) acc[t] = 0.0f;
}

// ---------------------------------------------------------------------------
// Kernel A: per-element BCE (masked, wave-reduced, atomics) + encode probs.
// enc[i] = valid ? sigmoid(x) : -1.0   (valid == !ignore). sigmoid >= 0 always,
// so sign encodes validity; one f32 stream feeds the whole phase-B gather.
// ---------------------------------------------------------------------------
__global__ __launch_bounds__(256)
void bce_enc_kernel(const float* __restrict__ logits,
                    const float* __restrict__ targets,
                    const unsigned char* __restrict__ sup,
                    const unsigned char* __restrict__ ign,
                    float* __restrict__ enc,
                    float* __restrict__ acc /* [0]=bce_sum [1]=bce_cnt */) {
    int idx = blockIdx.x * blockDim.x + threadIdx.x;
    float s = 0.0f, c = 0.0f;
    if (idx < B_ * N_) {
        float x = logits[idx];
        float t = targets[idx];
        bool is_sup = sup[idx] != 0;
        bool is_ign = ign[idx] != 0;
        // numerically stable BCE-with-logits
        float per = fmaxf(x, 0.0f) - x * t + log1pf(expf(-fabsf(x)));
        if (is_sup) { s = per; c = 1.0f; }
        float p = sigmoidf_(x);
        enc[idx] = is_ign ? -1.0f : p;
    }
    // EXEC is full here (guard above is uniform-exit, grid is an exact multiple)
    s = wave_red_add(s);
    c = wave_red_add(c);
    if ((threadIdx.x & 31) == 0) {
        atomicAdd(&acc[0], s);
        atomicAdd(&acc[1], c);
    }
}

// ---------------------------------------------------------------------------
// Kernel B: graph smoothness. One 128-thread block per (b, nb).
// Double-buffered async gather of neighbor blocks (512 B each) into LDS.
// ---------------------------------------------------------------------------
__global__ __launch_bounds__(BS_)
void graph_kernel(const float* __restrict__ logits,
                  const unsigned char* __restrict__ sup,
                  const unsigned char* __restrict__ ign,
                  const int* __restrict__ kv_indices,
                  const int* __restrict__ kv_num_blocks,
                  const float* __restrict__ enc,
                  float* __restrict__ loss_b,   // [B]
                  float* __restrict__ nunc_b) { // [B]
    const int blk = blockIdx.x;          // 0 .. B*NB-1
    const int b   = blk / NB_;
    const int nb  = blk - b * NB_;
    const int e   = threadIdx.x;         // 0 .. 127
    const int gbase = b * N_;
    const int gidx  = gbase + nb * BS_ + e;

    const float x   = logits[gidx];
    const float p   = sigmoidf_(x);
    const bool  unc = (ign[gidx] == 0) && (sup[gidx] == 0);
    const float uf  = unc ? 1.0f : 0.0f;

    const int num_raw = kv_num_blocks[blk];
    int num = num_raw < 0 ? 0 : (num_raw > K_ ? K_ : num_raw);
    const int* kvrow = kv_indices + (size_t)blk * K_;

    float acc_p = 0.0f, acc_c = 0.0f;
    const float* gsrc = enc + gbase + e; // + j*BS_ selects neighbor block

#if ASYNC_LDS
    // ---- CDNA5 path: double-buffered GLOBAL_LOAD_ASYNC_TO_LDS_B32 ----
    __shared__ float sbuf[2][BS_];
    typedef __attribute__((address_space(1))) int gas_int;
    typedef __attribute__((address_space(3))) int las_int;
    // Generic LDS address -> LDS byte offset = low 32 bits (ISA §10.2 aperture map)
    las_int* l0 = (las_int*)(uint32_t)(uintptr_t)(void*)&sbuf[0][e];
    las_int* l1 = (las_int*)(uint32_t)(uintptr_t)(void*)&sbuf[1][e];
    if (num > 0) {
        gas_int* g0 = (gas_int*)(uintptr_t)(const void*)(gsrc + kvrow[0] * BS_);
        __builtin_amdgcn_global_load_async_to_lds_b32(g0, l0, 0, 0);
    }
    for (int k = 0; k < num; ++k) {
        if (k + 1 < num) {
            las_int* ln = ((k + 1) & 1) ? l1 : l0;
            gas_int* gn = (gas_int*)(uintptr_t)(const void*)(gsrc + kvrow[k + 1] * BS_);
            __builtin_amdgcn_global_load_async_to_lds_b32(gn, ln, 0, 0);
            // async loads complete in order: <=1 outstanding means load k landed
#if __has_builtin(__builtin_amdgcn_s_wait_asynccnt)
            __builtin_amdgcn_s_wait_asynccnt(1);
#else
            asm volatile("s_wait_asynccnt 0x1" ::: "memory");
#endif
        } else {
#if __has_builtin(__builtin_amdgcn_s_wait_asynccnt)
            __builtin_amdgcn_s_wait_asynccnt(0);
#else
            asm volatile("s_wait_asynccnt 0x0" ::: "memory");
#endif
        }
        asm volatile("" ::: "memory"); // keep the LDS read after the wait
        float v = sbuf[k & 1][e];      // each lane reads only its own slot
        acc_p += fmaxf(v, 0.0f);            // p * valid
        acc_c += (v >= 0.0f) ? 1.0f : 0.0f; // valid count
    }
#else
    // ---- fallback: direct loads + global_prefetch_b8 of next block ----
    for (int k = 0; k < num; ++k) {
        const float* src = gsrc + kvrow[k] * BS_;
        if (k + 1 < num) __builtin_prefetch(gsrc + kvrow[k + 1] * BS_, 0, 0);
        float v = *src;
        acc_p += fmaxf(v, 0.0f);
        acc_c += (v >= 0.0f) ? 1.0f : 0.0f;
    }
#endif

    // block-level reduction (4 wave32 partials through LDS; per-wave sums on
    // the matrix pipe via V_WMMA_F32_16X16X4_F32)
    __shared__ float red[3][4];
    float rp = wave_red_add(acc_p);
    float rc = wave_red_add(acc_c);
    float ru = wave_red_add(uf);
    const int wave = threadIdx.x >> 5;
    if ((threadIdx.x & 31) == 0) { red[0][wave] = rp; red[1][wave] = rc; red[2][wave] = ru; }
    __syncthreads();
    float nsum = red[0][0] + red[0][1] + red[0][2] + red[0][3];
    float cnt  = red[1][0] + red[1][1] + red[1][2] + red[1][3];
    float qunc = red[2][0] + red[2][1] + red[2][2] + red[2][3];
    __syncthreads();

    float nmean = nsum / fmaxf(cnt, 1.0f);
    float d = p - nmean;
    float sqe = d * d * uf;
    float rs = wave_red_add(sqe);
    if ((threadIdx.x & 31) == 0) red[0][wave] = rs;
    __syncthreads();

    if (threadIdx.x == 0) {
        float sq = red[0][0] + red[0][1] + red[0][2] + red[0][3];
        bool ok = (qunc > 0.0f) && (num_raw > 0) && (cnt > 0.0f);
        if (ok) {
            atomicAdd(&loss_b[b], sq);
            atomicAdd(&nunc_b[b], qunc);
        }
    }
}

// ---------------------------------------------------------------------------
// Kernel C: finalize to scalar. Single wave32 (shuffle reduction is fine here).
// ---------------------------------------------------------------------------
__global__ __launch_bounds__(32)
void finalize_kernel(const float* __restrict__ acc,     // [0]=bce_sum [1]=bce_cnt
                     const float* __restrict__ loss_b,  // [B]
                     const float* __restrict__ nunc_b,  // [B]
                     float* __restrict__ out) {
    int t = threadIdx.x;
    float per = 0.0f, okf = 0.0f;
    if (t < B_) {
        float n = nunc_b[t];
        if (n > 0.0f) { per = loss_b[t] / fmaxf(n, 1.0f); okf = 1.0f; }
    }
    per = wave_red_shfl(per);
    okf = wave_red_shfl(okf);
    if (t == 0) {
        float graph = per / fmaxf(okf, 1.0f);
        float bce = acc[0] / fmaxf(acc[1], 1.0f);
        out[0] = bce + GRAPH_WEIGHT_ * graph;
    }
}

// ---------------------------------------------------------------------------
// launch: inputs in setup_inputs() order:
// 0 logits f32[B,N,1], 1 targets f32[B,N,1], 2 sup_mask bool[B,N],
// 3 ignore_mask bool[B,N], 4 kv_indices i32[B,NB,K], 5 kv_num_blocks i32[B,NB],
// 6 block_size scalar. bools assumed 1 byte (numpy bool_).
// Workspace: enc f32[B*N] (4 MB) followed by 34 accumulator floats.
// ---------------------------------------------------------------------------
extern "C" void kernel_launch(void* const* d_in, const int* in_sizes, int n_in,
                              void* d_out, int out_size, void* d_ws, size_t ws_size,
                              hipStream_t stream) {
    const float*         logits  = (const float*)d_in[0];
    const float*         targets = (const float*)d_in[1];
    const unsigned char* supm    = (const unsigned char*)d_in[2];
    const unsigned char* ignm    = (const unsigned char*)d_in[3];
    const int*           kvi     = (const int*)d_in[4];
    const int*           kvn     = (const int*)d_in[5];
    float*               out     = (float*)d_out;

    float* enc = (float*)d_ws;                 // B*N floats
    float* acc = enc + (size_t)B_ * N_;        // 34 floats: [bce_sum, bce_cnt, loss_b[16], nunc_b[16]]
    float* loss_b = acc + 2;
    float* nunc_b = acc + 18;

    zero_acc_kernel<<<1, 64, 0, stream>>>(acc);
    bce_enc_kernel<<<(B_ * N_ + 255) / 256, 256, 0, stream>>>(
        logits, targets, supm, ignm, enc, acc);
    graph_kernel<<<B_ * NB_, BS_, 0, stream>>>(
        logits, supm, ignm, kvi, kvn, enc, loss_b, nunc_b);
    finalize_kernel<<<1, 32, 0, stream>>>(acc, loss_b, nunc_b, out);
}